// ChamferLossKL_72688026517737
// MI455X (gfx1250) — compile-verified
//
#include <hip/hip_runtime.h>

// ---------------------------------------------------------------------------
// Chamfer-KL loss for Gaussians: bs=8, n=2048, d=4, fp32.
// Pairwise KL matrix expressed as a K=16 GEMM -> V_WMMA_F32_16X16X4_F32 x4.
// Feature vectors (per point, K index 0..15):
//   A_i = [0.5*exp(la_d) x4 | 0.5*mua_d^2 x4 | -mua_d x4 | 1, c_i, 0, 0]
//   B_j = [ivb_d x4        | ivb_d x4       | mub_d*ivb_d x4 | e_j, 1, 0, 0]
//   c_i = -0.5*(4 + sum_d la_i,d)
//   e_j =  0.5*(sum_d lb_j,d + sum_d mub_j,d^2 * ivb_j,d),  ivb = exp(-lb)
// Then p_kl[i,j] = dot(A_i, B_j).
// ---------------------------------------------------------------------------

typedef float v2f __attribute__((ext_vector_type(2)));
typedef float v8f __attribute__((ext_vector_type(8)));

#define NBATCH 8
#define NPTS   2048
#define NTILE  128                 // NPTS / 16
#define FEATF  (NBATCH * NTILE * 32 * 8)   // floats per feature array (1 MB)

// ---------------------------------------------------------------------------
// Kernel 1: build WMMA-layout feature fragments + init colmin + zero out.
// One thread per (b, point); tid == b*2048 + i, which is also the float4 index.
// Fragment layout: feat[((b*128 + tile)*32 + lane)*8 + slot]
//   lane = (i%16) + 16*hi ; slot pair (2c,2c+1) holds K = 4c + 2*hi + {0,1}
// ---------------------------------------------------------------------------
__global__ void kl_prep(const float* __restrict__ muA, const float* __restrict__ lvA,
                        const float* __restrict__ muB, const float* __restrict__ lvB,
                        float* __restrict__ Af, float* __restrict__ Bf,
                        float* __restrict__ colmin, float* __restrict__ out) {
  const int tid = blockIdx.x * blockDim.x + threadIdx.x;   // 0 .. 16383
  colmin[tid] = __builtin_inff();
  if (tid < NBATCH) out[tid] = 0.0f;

  const int b = tid >> 11;
  const int i = tid & (NPTS - 1);

  const float4 ma = ((const float4*)muA)[tid];
  const float4 la = ((const float4*)lvA)[tid];
  const float4 mb = ((const float4*)muB)[tid];
  const float4 lb = ((const float4*)lvB)[tid];

  const int rt = i >> 4;
  const int m  = i & 15;
  const int base = (((b << 7) + rt) << 5) + m;             // lane index (hi=0)

  // ---- A features (preds) ----
  const float ea0 = expf(la.x), ea1 = expf(la.y), ea2 = expf(la.z), ea3 = expf(la.w);
  const float ci  = -0.5f * (4.0f + la.x + la.y + la.z + la.w);
  float4* a0 = (float4*)(Af + (size_t)base * 8);
  float4* a1 = (float4*)(Af + (size_t)(base + 16) * 8);
  a0[0] = make_float4(0.5f * ea0, 0.5f * ea1, 0.5f * ma.x * ma.x, 0.5f * ma.y * ma.y);
  a0[1] = make_float4(-ma.x, -ma.y, 1.0f, ci);
  a1[0] = make_float4(0.5f * ea2, 0.5f * ea3, 0.5f * ma.z * ma.z, 0.5f * ma.w * ma.w);
  a1[1] = make_float4(-ma.z, -ma.w, 0.0f, 0.0f);

  // ---- B features (gts) ----
  const float iv0 = expf(-lb.x), iv1 = expf(-lb.y), iv2 = expf(-lb.z), iv3 = expf(-lb.w);
  const float ej = 0.5f * (lb.x + lb.y + lb.z + lb.w +
                           mb.x * mb.x * iv0 + mb.y * mb.y * iv1 +
                           mb.z * mb.z * iv2 + mb.w * mb.w * iv3);
  float4* g0 = (float4*)(Bf + (size_t)base * 8);
  float4* g1 = (float4*)(Bf + (size_t)(base + 16) * 8);
  g0[0] = make_float4(iv0, iv1, iv0, iv1);
  g0[1] = make_float4(mb.x * iv0, mb.y * iv1, ej, 1.0f);
  g1[0] = make_float4(iv2, iv3, iv2, iv3);
  g1[1] = make_float4(mb.z * iv2, mb.w * iv3, 0.0f, 0.0f);
}

// ---------------------------------------------------------------------------
// Kernel 2: one wave per (batch, 16-row tile). 4 waves / block, 256 blocks.
// Branch-free inner loop: 2x b128 load -> 4x v_wmma_f32_16x16x4_f32 ->
// min3 tree -> hardware float atomic-min (all 32 lanes; the two lane halves
// of a column combine at the L2 atomic unit).
// ---------------------------------------------------------------------------
__global__ void __launch_bounds__(128) kl_main(const float* __restrict__ Af,
                                               const float* __restrict__ Bf,
                                               float* __restrict__ colmin,
                                               float* __restrict__ out) {
  const int lane = threadIdx.x & 31;
  const int wave = threadIdx.x >> 5;
  const int tile = blockIdx.x * 4 + wave;          // 0 .. 1023
  const int b    = tile >> 7;
  const int rt   = tile & (NTILE - 1);

  // A fragment for this wave's 16 rows (held for the whole loop)
  const float4* Ap = (const float4*)(Af + (size_t)((((b << 7) + rt) << 5) + lane) * 8);
  const float4 a0 = Ap[0];
  const float4 a1 = Ap[1];
  const v2f aC0 = {a0.x, a0.y};
  const v2f aC1 = {a0.z, a0.w};
  const v2f aC2 = {a1.x, a1.y};
  const v2f aC3 = {a1.z, a1.w};

  float* cmbase = colmin + (b << 11) + (lane & 15);

  v8f rmin;
#pragma unroll
  for (int s = 0; s < 8; ++s) rmin[s] = __builtin_inff();

  for (int ct = 0; ct < NTILE; ++ct) {
    const float4* Bp = (const float4*)(Bf + (size_t)((((b << 7) + ct) << 5) + lane) * 8);
    const float4 b0 = Bp[0];
    const float4 b1 = Bp[1];
    const v2f bC0 = {b0.x, b0.y};
    const v2f bC1 = {b0.z, b0.w};
    const v2f bC2 = {b1.x, b1.y};
    const v2f bC3 = {b1.z, b1.w};

    v8f acc = {};
    acc = __builtin_amdgcn_wmma_f32_16x16x4_f32(false, aC0, false, bC0, (short)0, acc, false, false);
    acc = __builtin_amdgcn_wmma_f32_16x16x4_f32(false, aC1, false, bC1, (short)0, acc, false, false);
    acc = __builtin_amdgcn_wmma_f32_16x16x4_f32(false, aC2, false, bC2, (short)0, acc, false, false);
    acc = __builtin_amdgcn_wmma_f32_16x16x4_f32(false, aC3, false, bC3, (short)0, acc, false, false);

    // running row mins + this tile's per-lane column min (8 rows of one half)
    float cmin = fminf(acc[0], acc[1]);
#pragma unroll
    for (int s = 0; s < 8; ++s) {
      rmin[s] = fminf(rmin[s], acc[s]);
      if (s >= 2) cmin = fminf(cmin, acc[s]);
    }
    // both lane halves hit the same address -> full 16-row min in memory
    (void)__hip_atomic_fetch_min(cmbase + (ct << 4), cmin,
                                 __ATOMIC_RELAXED, __HIP_MEMORY_SCOPE_AGENT);
  }

  // reduce row mins across the 16 lanes of each half (N dimension)
#pragma unroll
  for (int msk = 8; msk; msk >>= 1) {
#pragma unroll
    for (int s = 0; s < 8; ++s)
      rmin[s] = fminf(rmin[s], __shfl_xor(rmin[s], msk, 32));
  }
  float halfsum = 0.0f;
#pragma unroll
  for (int s = 0; s < 8; ++s) halfsum += rmin[s];
  // lane 0 holds rows 0..7 of this tile, lane 16 holds rows 8..15
  if (lane == 0 || lane == 16) atomicAdd(&out[b], halfsum);   // loss_2 part
}

// ---------------------------------------------------------------------------
// Kernel 3: per-batch sum of column mins (loss_1), added into out[b].
// ---------------------------------------------------------------------------
__global__ void kl_final(const float* __restrict__ colmin, float* __restrict__ out) {
  __shared__ float sd[256];
  const int b = blockIdx.x;
  float s = 0.0f;
  for (int j = threadIdx.x; j < NPTS; j += 256)
    s += colmin[(b << 11) + j];
  sd[threadIdx.x] = s;
  __syncthreads();
  for (int st = 128; st; st >>= 1) {
    if (threadIdx.x < st) sd[threadIdx.x] += sd[threadIdx.x + st];
    __syncthreads();
  }
  if (threadIdx.x == 0) out[b] += sd[0];
}

// ---------------------------------------------------------------------------
extern "C" void kernel_launch(void* const* d_in, const int* in_sizes, int n_in,
                              void* d_out, int out_size, void* d_ws, size_t ws_size,
                              hipStream_t stream) {
  const float* mu_preds     = (const float*)d_in[0];
  const float* logvar_preds = (const float*)d_in[1];
  const float* mu_gts       = (const float*)d_in[2];
  const float* logvar_gts   = (const float*)d_in[3];
  float* out = (float*)d_out;

  float* Af     = (float*)d_ws;                 // 1 MB: A feature fragments
  float* Bf     = Af + FEATF;                   // 1 MB: B feature fragments
  float* colmin = Bf + FEATF;                   // 64 KB: per-column mins

  // 1) features + init (16384 threads)
  kl_prep<<<NBATCH * NPTS / 256, 256, 0, stream>>>(mu_preds, logvar_preds,
                                                   mu_gts, logvar_gts,
                                                   Af, Bf, colmin, out);
  // 2) WMMA pairwise + mins (1024 waves = 256 blocks x 4 waves)
  kl_main<<<NBATCH * NTILE / 4, 128, 0, stream>>>(Af, Bf, colmin, out);
  // 3) loss_1 accumulation
  kl_final<<<NBATCH, 256, 0, stream>>>(colmin, out);
}